// System1LoRA_71287867179044
// MI455X (gfx1250) — compile-verified
//
#include <hip/hip_runtime.h>
#include <math.h>

// ---------------------------------------------------------------------------
// System1LoRA recurrence on MI455X (gfx1250, wave32, WMMA + TDM).
// 10 sequential steps; per step:
//   h_n = h + 0.01*noise[t]                          (k_hn -> hcat[:, :H])
//   x_logits = h_n @ W_x + b_x                       (k_gemm_wmma, f32 WMMA)
//   op_logits = h_n @ W_op + b_op ; all-stop flag    (k_op, W_op staged via TDM)
//   ptr = first-argmax(x_logits) ; gather s2a row    (k_ptr_gather -> hcat[:, H:])
//   h_new = hcat @ W_res + b_res                     (k_gemm_wmma, f32 WMMA)
//   h = done ? h : h_new                             (k_freeze, old done)
//   done |= all_stop                                 (k_update_done)
// ---------------------------------------------------------------------------

typedef __attribute__((ext_vector_type(2))) float v2f;
typedef __attribute__((ext_vector_type(8))) float v8f;
typedef __attribute__((ext_vector_type(4))) unsigned int u32x4;
typedef __attribute__((ext_vector_type(8))) int i32x8;
typedef __attribute__((ext_vector_type(4))) int i32x4;

#define B_   64
#define S_   4096
#define H_   1024
#define KTILE 64

// ------------------------- init: h = 0, done = 0 ---------------------------
__global__ void k_init(float* __restrict__ h, int* __restrict__ done) {
    int idx = blockIdx.x * 256 + threadIdx.x;      // B*H = 65536 threads
    h[idx] = 0.0f;
    if (idx == 0) *done = 0;
}

// --------------------- h_n = h + 0.01*noise -> hcat[:, :H] -----------------
__global__ void k_hn(const float* __restrict__ h, const float* __restrict__ noise_t,
                     float* __restrict__ hcat) {
    int idx = blockIdx.x * 256 + threadIdx.x;      // 0 .. B*H-1
    int b = idx >> 10;                             // /H_
    int j = idx & (H_ - 1);
    hcat[(size_t)b * (2 * H_) + j] = h[idx] + 0.01f * noise_t[idx];
}

// ------------------ generic f32 WMMA GEMM: C = A@W + bias ------------------
// M fixed = 64. Block = 128 threads (4 waves); wave w -> rows [16w, 16w+16).
// blockIdx.x -> 16-col N tile. K looped in KTILE chunks staged in LDS.
__global__ void k_gemm_wmma(const float* __restrict__ A, int lda,
                            const float* __restrict__ W,
                            const float* __restrict__ bias,
                            float* __restrict__ C, int N, int K) {
    __shared__ __align__(16) float Bt[KTILE * 16];

    const int tid  = threadIdx.x;
    const int lane = tid & 31;
    const int wave = tid >> 5;
    const int half = lane >> 4;       // 0: K+{0,1}, 1: K+{2,3}
    const int l15  = lane & 15;
    const int n0   = blockIdx.x * 16;
    const int m0   = wave * 16;

    v8f acc = {};

    for (int k0 = 0; k0 < K; k0 += KTILE) {
        // Cooperative stage of W[k0..k0+63][n0..n0+15] (1024 floats, 256 x f4)
        #pragma unroll
        for (int i = 0; i < (KTILE * 16) / (128 * 4); ++i) {
            int idx = tid + i * 128;              // float4 index
            int row = idx >> 2;
            int seg = idx & 3;
            const float4 v = *reinterpret_cast<const float4*>(
                &W[(size_t)(k0 + row) * N + n0 + seg * 4]);
            *reinterpret_cast<float4*>(&Bt[row * 16 + seg * 4]) = v;
            if (k0 + KTILE < K)                   // prefetch next chunk's tile
                __builtin_prefetch(&W[(size_t)(k0 + KTILE + row) * N + n0 + seg * 4], 0, 0);
        }
        __syncthreads();

        const float* Arow = &A[(size_t)(m0 + l15) * lda + k0 + half * 2];
        #pragma unroll
        for (int kk = 0; kk < KTILE; kk += 4) {
            // A fragment (16x4 f32): lane L holds row L&15, K = kk + half*2 {+1}
            v2f a = *reinterpret_cast<const v2f*>(Arow + kk);
            // B fragment (4x16 f32): lane holds col l15, K rows half*2 {+1}
            v2f b;
            b.x = Bt[(kk + half * 2 + 0) * 16 + l15];
            b.y = Bt[(kk + half * 2 + 1) * 16 + l15];
            acc = __builtin_amdgcn_wmma_f32_16x16x4_f32(
                false, a, false, b, (short)0, acc, false, false);
        }
        __syncthreads();
    }

    const float bv = bias[n0 + l15];
    #pragma unroll
    for (int r = 0; r < 8; ++r) {
        int row = m0 + r + half * 8;              // C/D layout: M=r and M=r+8
        C[(size_t)row * N + n0 + l15] = acc[r] + bv;
    }
}

// ------ op_logits = h_n @ W_op + b_op ; all_stop = all(argmax == 0) --------
// W_op (32 KB) is staged into LDS with one Tensor-Data-Mover descriptor.
__global__ void k_op(const float* __restrict__ hcat, const float* __restrict__ Wop,
                     const float* __restrict__ bop, int* __restrict__ all_stop) {
    __shared__ __align__(16) float wop[H_ * 8];   // 32 KB
    __shared__ float opl[B_ * 8];
    __shared__ int   sAll;

    int tid = threadIdx.x;                        // 256 threads, 1 block
    if (tid == 0) sAll = 1;

#if defined(__gfx1250__)
    if (tid < 32) {                               // wave 0 issues TDM load
        unsigned int lds_off = (unsigned int)(size_t)(void*)&wop[0];
        unsigned long long ga = (unsigned long long)(size_t)Wop;
        const unsigned int td0 = H_ * 8;          // 8192 f32 elements (1-D tile)
        u32x4 g0;
        g0[0] = 1u;                               // count=1, user descriptor
        g0[1] = lds_off;                          // lds_addr (bytes)
        g0[2] = (unsigned int)ga;                 // global_addr[31:0]
        g0[3] = (unsigned int)((ga >> 32) & 0x01FFFFFFu) | (2u << 30); // [56:32]|type=2
        i32x8 g1;
        g1[0] = (int)(2u << 16);                  // data_size = 4 bytes
        g1[1] = (int)((td0 & 0xFFFFu) << 16);     // tensor_dim0[15:0]
        g1[2] = (int)((td0 >> 16) | (1u << 16));  // tensor_dim0[31:16] | tensor_dim1=1
        g1[3] = (int)((td0 & 0xFFFFu) << 16);     // tile_dim0 = 8192
        g1[4] = 0;                                // tile_dim1/2 unused
        g1[5] = (int)td0;                         // tensor_dim0_stride[31:0]
        g1[6] = 0;
        g1[7] = 0;
        i32x4 z4 = {0, 0, 0, 0};
  #if __clang_major__ >= 23
        i32x8 z8 = {0, 0, 0, 0, 0, 0, 0, 0};
        __builtin_amdgcn_tensor_load_to_lds(g0, g1, z4, z4, z8, 0);
  #else
        __builtin_amdgcn_tensor_load_to_lds(g0, g1, z4, z4, 0);
  #endif
        __builtin_amdgcn_s_wait_tensorcnt(0);
    }
#else
    for (int i = tid; i < H_ * 8; i += 256) wop[i] = Wop[i];
#endif
    __syncthreads();

    for (int p = tid; p < B_ * 8; p += 256) {     // 512 (row,op) pairs
        int row = p >> 3, op = p & 7;
        const float* a = &hcat[(size_t)row * (2 * H_)];
        float s = 0.0f;
        for (int k = 0; k < H_; ++k) s += a[k] * wop[k * 8 + op];
        opl[p] = s + bop[op];
    }
    __syncthreads();

    if (tid < B_) {
        float best = opl[tid * 8];
        int bi = 0;
        #pragma unroll
        for (int o = 1; o < 8; ++o) {             // first-occurrence argmax
            float v = opl[tid * 8 + o];
            if (v > best) { best = v; bi = o; }
        }
        if (bi != 0) atomicAnd(&sAll, 0);
    }
    __syncthreads();
    if (tid == 0) *all_stop = sAll;
}

// ---- ptr = first-argmax over x_logits row; gather s2a row -> hcat[:, H:] --
__global__ void k_ptr_gather(const float* __restrict__ xl,
                             const float* __restrict__ s2a,
                             float* __restrict__ hcat) {
    __shared__ float sv[256];
    __shared__ int   si[256];
    int b = blockIdx.x, tid = threadIdx.x;

    const float* row = &xl[(size_t)b * S_];
    float best = -INFINITY;
    int   bi   = 0;
    for (int i = tid; i < S_; i += 256) {
        float v = row[i];
        if (v > best || (v == best && i < bi)) { best = v; bi = i; }
    }
    sv[tid] = best; si[tid] = bi;
    __syncthreads();
    for (int s = 128; s > 0; s >>= 1) {
        if (tid < s) {
            float v2 = sv[tid + s]; int i2 = si[tid + s];
            if (v2 > sv[tid] || (v2 == sv[tid] && i2 < si[tid])) {
                sv[tid] = v2; si[tid] = i2;
            }
        }
        __syncthreads();
    }
    int ptr = si[0];                              // < S_, min(.,S-1) is no-op
    const float4* src4 = reinterpret_cast<const float4*>(
        &s2a[((size_t)b * S_ + ptr) * H_]);
    float4* dst4 = reinterpret_cast<float4*>(&hcat[(size_t)b * (2 * H_) + H_]);
    dst4[tid] = src4[tid];                        // 256 * float4 = 1024 floats
}

// -------------------- h = done ? h : h_new (old done) ----------------------
__global__ void k_freeze(float* __restrict__ h, const float* __restrict__ h_new,
                         const int* __restrict__ done) {
    int idx = blockIdx.x * 256 + threadIdx.x;
    if (!(*done)) h[idx] = h_new[idx];
}

// ------------------------- done |= all_stop --------------------------------
__global__ void k_update_done(int* __restrict__ done, const int* __restrict__ all_stop) {
    *done = *done | *all_stop;
}

// ---------------------------------------------------------------------------
extern "C" void kernel_launch(void* const* d_in, const int* in_sizes, int n_in,
                              void* d_out, int out_size, void* d_ws, size_t ws_size,
                              hipStream_t stream) {
    const float* s2a   = (const float*)d_in[0];   // [B,S,H]
    const float* Wres  = (const float*)d_in[1];   // [2H,H]
    const float* bres  = (const float*)d_in[2];   // [H]
    const float* Wop   = (const float*)d_in[3];   // [H,8]
    const float* bop   = (const float*)d_in[4];   // [8]
    const float* Wx    = (const float*)d_in[5];   // [H,S]
    const float* bx    = (const float*)d_in[6];   // [S]
    const float* noise = (const float*)d_in[7];   // [10,B,H]

    float* h = (float*)d_out;                     // [B,H] lives in d_out

    float* hcat = (float*)d_ws;                   // [B, 2H]   = 131072 f
    float* xl   = hcat + (size_t)B_ * 2 * H_;     // [B, S]    = 262144 f
    float* hnew = xl + (size_t)B_ * S_;           // [B, H]    =  65536 f
    int*   done = (int*)(hnew + (size_t)B_ * H_);
    int*   allst = done + 1;

    k_init<<<256, 256, 0, stream>>>(h, done);

    for (int t = 0; t < 10; ++t) {
        const float* noise_t = noise + (size_t)t * B_ * H_;
        k_hn<<<256, 256, 0, stream>>>(h, noise_t, hcat);
        // x_logits = h_n @ W_x + b_x   (M=64,K=1024,N=4096)
        k_gemm_wmma<<<S_ / 16, 128, 0, stream>>>(hcat, 2 * H_, Wx, bx, xl, S_, H_);
        k_op<<<1, 256, 0, stream>>>(hcat, Wop, bop, allst);
        k_ptr_gather<<<B_, 256, 0, stream>>>(xl, s2a, hcat);
        // h_new = [h_n|borrowed] @ W_res + b_res  (M=64,K=2048,N=1024)
        k_gemm_wmma<<<H_ / 16, 128, 0, stream>>>(hcat, 2 * H_, Wres, bres, hnew, H_, 2 * H_);
        k_freeze<<<256, 256, 0, stream>>>(h, hnew, done);
        k_update_done<<<1, 1, 0, stream>>>(done, allst);
    }
}